// Get_gradient_loss_10591389352533
// MI455X (gfx1250) — compile-verified
//
#include <hip/hip_runtime.h>
#include <hip/hip_bf16.h>

typedef float v2f __attribute__((ext_vector_type(2)));
typedef float v8f __attribute__((ext_vector_type(8)));

#define NDIM   256
#define NN     (NDIM * NDIM)          // 65536
#define NNN    (NDIM * NDIM * NDIM)   // 16777216
#define EPS_F  1e-6f

#define BLOCKS   8192
#define THREADS  256
#define TOTAL_T  (BLOCKS * THREADS)   // 2097152; NNN / TOTAL_T == 8 exactly

// One-instruction XOR-butterfly stage: ds_swizzle_b32 group-of-32 mode.
// offset[14:10] = xor_mask, offset[9:5] = or_mask(0), offset[4:0] = and_mask(0x1f).
#define SWZ_XOR_ADD(t, xmask)                                                  \
  ((t) + __int_as_float(__builtin_amdgcn_ds_swizzle(                           \
             __float_as_int(t), (((xmask) << 10) | 0x1f))))

// Gradient magnitude at clamped center (dc,hc,wc), all in [1,254].
// D-axis stride = NN, H-axis stride = NDIM, W-axis stride = 1.
__device__ __forceinline__ float grad_mag(const float* __restrict__ x,
                                          int dc, int hc, int wc) {
  const int base = dc * NN + hc * NDIM + wc;
  const float gw = x[base + NN]   - x[base - NN];   // along D
  const float gd = x[base + NDIM] - x[base - NDIM]; // along H
  const float gh = x[base + 1]    - x[base - 1];    // along W
  // Raw v_sqrt_f32 (~1 ulp): args are >= EPS so no special-case handling needed.
  return __builtin_amdgcn_sqrtf(gw * gw + gd * gd + gh * gh + EPS_F);
}

// Wave32 sum of per-lane value p via V_WMMA_F32_16X16X4_F32 with all-ones A.
// D[i][j] = sum_k B[k][j]; rows of D are identical, and VGPR0 of D carries rows
// M=0 and M=8, so summing d[0] across all 32 lanes yields 2 * sum(B elements)
// regardless of the exact hardware B layout. Requires EXEC all-1s.
__device__ __forceinline__ float wave_sum_wmma(float p) {
  v2f a; a.x = 1.0f; a.y = 1.0f;   // entire 16x4 A matrix = 1.0
  v2f b; b.x = p;    b.y = 0.0f;   // 32 partials + 32 zeros fill 4x16 B
  v8f c = {};
  c = __builtin_amdgcn_wmma_f32_16x16x4_f32(false, a, false, b,
                                            (short)0, c, false, false);
  float t = c[0];
  t = SWZ_XOR_ADD(t, 16);
  t = SWZ_XOR_ADD(t, 8);
  t = SWZ_XOR_ADD(t, 4);
  t = SWZ_XOR_ADD(t, 2);
  t = SWZ_XOR_ADD(t, 1);
  return t * 0.5f;
}

// Block-level sum (result valid in thread 0).
__device__ __forceinline__ float block_sum(float p) {
  __shared__ float wsum[THREADS / 32];
  const float t = wave_sum_wmma(p);
  const unsigned lane = threadIdx.x & 31u;
  const unsigned wave = threadIdx.x >> 5;
  if (lane == 0) wsum[wave] = t;
  __syncthreads();
  float s = 0.0f;
  if (threadIdx.x == 0) {
#pragma unroll
    for (int i = 0; i < THREADS / 32; ++i) s += wsum[i];
  }
  return s;
}

__global__ __launch_bounds__(THREADS)
void grad_loss_partial(const float* __restrict__ x1,
                       const float* __restrict__ x2,
                       float* __restrict__ partial) {
  const unsigned tid0 = blockIdx.x * THREADS + threadIdx.x;
  float acc = 0.0f;
  // Exactly 8 full iterations for every thread: no divergence around WMMA.
#pragma unroll
  for (int it = 0; it < NNN / TOTAL_T; ++it) {
    const unsigned idx = tid0 + (unsigned)it * TOTAL_T;
    const int w = (int)(idx & 255u);          // fastest dim -> coalesced lanes
    const int h = (int)((idx >> 8) & 255u);
    const int d = (int)(idx >> 16);
    // Edge-replicate padding == clamp the gradient sample point to [1,254].
    const int wc = min(max(w, 1), NDIM - 2);
    const int hc = min(max(h, 1), NDIM - 2);
    const int dc = min(max(d, 1), NDIM - 2);
    const float m1 = grad_mag(x1, dc, hc, wc);
    const float m2 = grad_mag(x2, dc, hc, wc);
    acc += __builtin_fabsf(m1 - m2);
  }
  const float s = block_sum(acc);
  if (threadIdx.x == 0) partial[blockIdx.x] = s;
}

__global__ __launch_bounds__(THREADS)
void grad_loss_final(const float* __restrict__ partial,
                     float* __restrict__ out) {
  float acc = 0.0f;
  // 8192 partials = 256 threads * 32 strided, coalesced reads.
#pragma unroll
  for (int i = 0; i < BLOCKS / THREADS; ++i)
    acc += partial[threadIdx.x + (unsigned)i * THREADS];
  const float s = block_sum(acc);
  if (threadIdx.x == 0) out[0] = s * (1.0f / (float)NNN);
}

extern "C" void kernel_launch(void* const* d_in, const int* in_sizes, int n_in,
                              void* d_out, int out_size, void* d_ws, size_t ws_size,
                              hipStream_t stream) {
  const float* x1 = (const float*)d_in[0];
  const float* x2 = (const float*)d_in[1];
  float* partial = (float*)d_ws;            // 8192 floats = 32 KB scratch
  float* out = (float*)d_out;               // single fp32 scalar

  grad_loss_partial<<<BLOCKS, THREADS, 0, stream>>>(x1, x2, partial);
  grad_loss_final<<<1, THREADS, 0, stream>>>(partial, out);
}